// EigenSpace_38766374814279
// MI455X (gfx1250) — compile-verified
//
#include <hip/hip_runtime.h>
#include <cstdint>

// ---------------------------------------------------------------------------
// out[b, m] = sum_k U[m,k] * (L[k]*style[b,k]) + mu[m]
//   M = 64*64*32 = 131072, K = 512, N = 32.
// Memory-bound: streams 268 MB of fp32 U once -> ~12.2 us floor @ 23.3 TB/s.
// fp32 WMMA (V_WMMA_F32_16X16X4_F32) needed to stay under the memory bound.
// ---------------------------------------------------------------------------

typedef __attribute__((ext_vector_type(2))) float v2f;
typedef __attribute__((ext_vector_type(8))) float v8f;

#define DIM    512
#define NBATCH 32
#define MROWS  (64 * 64 * 32)     // 131072
#define MTILE  128                // rows per block (8 waves * 16)
#define KC     16                 // K chunk per stage
#define NSTAGE (DIM / KC)         // 32
#define UPAD   20                 // u_lds row stride (floats): 16B-aligned rows,
                                  // bank-conflict-free A-fragment ds_load_b64
#define UBUF   (MTILE * UPAD)     // floats per double-buffer half (2560)

// Async DMA: global -> LDS, 16 bytes per lane. Inline asm (portable across
// ROCm7.2 / amdgpu-toolchain; bypasses builtin-arity differences).
__device__ __forceinline__ void async_g2l_b128(unsigned lds_byte_off,
                                               const float* gptr) {
    asm volatile("global_load_async_to_lds_b128 %0, %1, off"
                 :: "v"(lds_byte_off), "v"((uint64_t)(uintptr_t)gptr)
                 : "memory");
}

__global__ __launch_bounds__(256, 1)
void eigenspace_wmma_f32(const float* __restrict__ style,   // [32, 512]
                         const float* __restrict__ U,       // [131072, 512]
                         const float* __restrict__ L,       // [512]
                         const float* __restrict__ mu,      // [131072]
                         float* __restrict__ out)           // [32, 131072]
{
    // S packed as k-pairs: Sp[kp][n] = (S[2kp][n], S[2kp+1][n]), kp=0..255, n=0..31
    __shared__ float s_lds[(DIM / 2) * NBATCH * 2];   // 16384 floats = 64 KB
    __shared__ float u_lds[2 * UBUF];                 // 5120 floats  = 20 KB

    const int tid    = threadIdx.x;
    const int wave   = tid >> 5;
    const int lane   = tid & 31;
    const int lane15 = lane & 15;
    const int klh    = lane >> 4;            // K-half select (A/B fragment layout)
    const int m0     = blockIdx.x * MTILE;

    const float*  Ub     = U + (size_t)m0 * DIM;
    const unsigned u_base = (unsigned)(uintptr_t)(&u_lds[0]);

    // ---- prefetch stage 0 of the U tile (async DMA to LDS) -----------------
    {
#pragma unroll
        for (int i = 0; i < 2; ++i) {
            const int c   = tid + i * 256;        // 512 chunks of 16B per stage
            const int row = c >> 2, seg = c & 3;
            async_g2l_b128(u_base + (unsigned)(row * UPAD * 4 + seg * 16),
                           Ub + (size_t)row * DIM + seg * 4);
        }
    }

    // ---- build S = L * style in LDS (coalesced reads; done once per block) -
    for (int idx = tid; idx < DIM * NBATCH; idx += 256) {
        const int n = idx >> 9;          // batch
        const int k = idx & (DIM - 1);
        const float v = L[k] * style[n * DIM + k];
        s_lds[(k >> 1) * (NBATCH * 2) + n * 2 + (k & 1)] = v;
    }

    // ---- main K loop: double-buffered async staging + fp32 WMMA ------------
    v8f acc0 = {0.f, 0.f, 0.f, 0.f, 0.f, 0.f, 0.f, 0.f};   // batches  0..15
    v8f acc1 = {0.f, 0.f, 0.f, 0.f, 0.f, 0.f, 0.f, 0.f};   // batches 16..31

    const int rowl = wave * 16 + lane15;      // this lane's U row in the tile

    for (int s = 0; s < NSTAGE; ++s) {
        if (s + 1 < NSTAGE) {
            const int buf1 = (s + 1) & 1;
#pragma unroll
            for (int i = 0; i < 2; ++i) {
                const int c   = tid + i * 256;
                const int row = c >> 2, seg = c & 3;
                async_g2l_b128(u_base + (unsigned)(buf1 * UBUF * 4 +
                                                   row * UPAD * 4 + seg * 16),
                               Ub + (size_t)row * DIM + (s + 1) * KC + seg * 4);
            }
            // stage s complete (async loads retire in order), stage s+1 in flight
            asm volatile("s_wait_asynccnt 0x2" ::: "memory");
        } else {
            asm volatile("s_wait_asynccnt 0x0" ::: "memory");
        }
        __syncthreads();   // stage-s tile visible to all waves

        const int buf = s & 1;
        const float* ap = &u_lds[buf * UBUF + rowl * UPAD + klh * 2];
        const float* bp = &s_lds[(s * 8 + klh) * (NBATCH * 2) + lane15 * 2];

#pragma unroll
        for (int ks = 0; ks < 4; ++ks) {       // 4 k-steps of K=4 per stage
            // A 16x4 fp32 fragment: lane m holds U[row][k0+2*klh], [k0+2*klh+1]
            v2f a  = *(const v2f*)(ap + ks * 4);
            // B 4x16 fp32 fragment: lane n holds S[k0+2*klh][n], S[k0+2*klh+1][n]
            v2f b0 = *(const v2f*)(bp + ks * 2 * (NBATCH * 2));
            v2f b1 = *(const v2f*)(bp + ks * 2 * (NBATCH * 2) + NBATCH);

            acc0 = __builtin_amdgcn_wmma_f32_16x16x4_f32(
                       false, a, false, b0, (short)0, acc0, false, false);
            acc1 = __builtin_amdgcn_wmma_f32_16x16x4_f32(
                       false, a, false, b1, (short)0, acc1, false, false);
        }
        __syncthreads();   // all waves done reading buf before it is refilled
    }

    // ---- epilogue: D[i] = row (klh*8 + i), col = lane15; add mu, store -----
    const int mb = m0 + wave * 16 + klh * 8;          // global row base (mult of 8)
    const float4 mu0 = *(const float4*)(mu + mb);
    const float4 mu1 = *(const float4*)(mu + mb + 4);

    {
        float* o = out + (size_t)lane15 * MROWS + mb;
        float4 r0 = make_float4(acc0[0] + mu0.x, acc0[1] + mu0.y,
                                acc0[2] + mu0.z, acc0[3] + mu0.w);
        float4 r1 = make_float4(acc0[4] + mu1.x, acc0[5] + mu1.y,
                                acc0[6] + mu1.z, acc0[7] + mu1.w);
        *(float4*)(o)     = r0;
        *(float4*)(o + 4) = r1;
    }
    {
        float* o = out + (size_t)(lane15 + 16) * MROWS + mb;
        float4 r0 = make_float4(acc1[0] + mu0.x, acc1[1] + mu0.y,
                                acc1[2] + mu0.z, acc1[3] + mu0.w);
        float4 r1 = make_float4(acc1[4] + mu1.x, acc1[5] + mu1.y,
                                acc1[6] + mu1.z, acc1[7] + mu1.w);
        *(float4*)(o)     = r0;
        *(float4*)(o + 4) = r1;
    }
}

extern "C" void kernel_launch(void* const* d_in, const int* in_sizes, int n_in,
                              void* d_out, int out_size, void* d_ws, size_t ws_size,
                              hipStream_t stream) {
    (void)in_sizes; (void)n_in; (void)out_size; (void)d_ws; (void)ws_size;
    const float* style = (const float*)d_in[0];   // [32, 512]
    const float* U     = (const float*)d_in[1];   // [64, 64, 32, 512]
    const float* L     = (const float*)d_in[2];   // [512]
    const float* mu    = (const float*)d_in[3];   // [64, 64, 32]
    float*       out   = (float*)d_out;           // [32, 64, 64, 32]

    dim3 grid(MROWS / MTILE);   // 1024 blocks
    dim3 block(256);            // 8 wave32
    eigenspace_wmma_f32<<<grid, block, 0, stream>>>(style, U, L, mu, out);
}